// DataEncoder_52716428591244
// MI455X (gfx1250) — compile-verified
//
#include <hip/hip_runtime.h>
#include <stdint.h>

#define STRIDE_   8
#define ANCHOR_WH 32.0f
#define CLS_TH    0.5f
#define NMS_TH    0.5f
#define EPS_      1e-9f
#define NWP       384     // mask row pitch in 32-bit words (covers N <= 12288)

typedef unsigned long long u64;

// ---------------------------------------------------------------------------
// Kernel 1: decode boxes + sigmoid(max logit) score + 64-bit sort keys.
// key = ((0xFFFFFFFF - float_bits(score)) << 32) | original_index
// ascending sort of key == stable descending sort by score.
// ---------------------------------------------------------------------------
__global__ void decode_kernel(const float* __restrict__ loc,
                              const float* __restrict__ cls,
                              const int* __restrict__ pW,
                              const int* __restrict__ pH,
                              float4* __restrict__ boxes,
                              u64* __restrict__ keys,
                              int N) {
    int i = blockIdx.x * blockDim.x + threadIdx.x;
    if (i >= N) return;

    int W = *pW, H = *pH;
    int fmw = (W + STRIDE_ - 1) / STRIDE_;
    float gw = (float)W / (float)fmw;
    float gh = (float)H / (float)((H + STRIDE_ - 1) / STRIDE_);

    int row = i / fmw;
    int col = i - row * fmw;
    float cx = (col + 0.5f) * gw;
    float cy = (row + 0.5f) * gh;

    float4 l = ((const float4*)loc)[i];
    float x = l.x * ANCHOR_WH + cx;
    float y = l.y * ANCHOR_WH + cy;
    float w = expf(l.z) * ANCHOR_WH;
    float h = expf(l.w) * ANCHOR_WH;

    float4 b;
    b.x = x - 0.5f * w;
    b.y = y - 0.5f * h;
    b.z = x + 0.5f * w;
    b.w = y + 0.5f * h;
    boxes[i] = b;

    const float4* c = (const float4*)(cls + (size_t)i * 64);
    float m = -INFINITY;
#pragma unroll
    for (int k = 0; k < 16; ++k) {
        float4 v = c[k];
        m = fmaxf(m, fmaxf(fmaxf(v.x, v.y), fmaxf(v.z, v.w)));
    }
    float score = 1.0f / (1.0f + expf(-m));   // in (0,1): positive float bits

    unsigned hi = 0xFFFFFFFFu - __float_as_uint(score);
    keys[i] = ((u64)hi << 32) | (unsigned)i;
}

// ---------------------------------------------------------------------------
// Kernel 2: single-workgroup bitonic sort of M (pow2 >= N) keys in LDS.
// 16384 * 8B = 128KB LDS -- CDNA5's 320KB/WGP makes this a one-WGP sort.
// ---------------------------------------------------------------------------
__global__ __launch_bounds__(1024) void sort_kernel(u64* __restrict__ keys,
                                                    int N, int M) {
    extern __shared__ char dyn_smem[];
    u64* sk = (u64*)dyn_smem;

    for (int s = threadIdx.x; s < M; s += blockDim.x)
        sk[s] = (s < N) ? keys[s] : ~0ull;
    __syncthreads();

    for (int k = 2; k <= M; k <<= 1) {
        for (int j = k >> 1; j > 0; j >>= 1) {
            for (int i = threadIdx.x; i < M; i += blockDim.x) {
                int ixj = i ^ j;
                if (ixj > i) {
                    u64 a = sk[i], b = sk[ixj];
                    bool up = ((i & k) == 0);
                    if ((a > b) == up) { sk[i] = b; sk[ixj] = a; }
                }
            }
            __syncthreads();
        }
    }

    for (int s = threadIdx.x; s < N; s += blockDim.x)
        keys[s] = sk[s];
}

// ---------------------------------------------------------------------------
// Kernel 3a: materialize boxes in sorted order (gather).
// ---------------------------------------------------------------------------
__global__ void gather_kernel(const u64* __restrict__ keys,
                              const float4* __restrict__ boxes,
                              float4* __restrict__ sboxes, int N) {
    int s = blockIdx.x * blockDim.x + threadIdx.x;
    if (s >= N) return;
    sboxes[s] = boxes[(unsigned)keys[s]];
}

// ---------------------------------------------------------------------------
// Kernel 3b: build validity bitmask words (score > 0.5) in sorted order.
// ---------------------------------------------------------------------------
__global__ void validw_kernel(const u64* __restrict__ keys,
                              unsigned* __restrict__ validw, int N) {
    int w = blockIdx.x * blockDim.x + threadIdx.x;
    if (w >= NWP) return;
    unsigned word = 0u;
    for (int b = 0; b < 32; ++b) {
        int s = (w << 5) + b;
        if (s < N) {
            unsigned sbits = 0xFFFFFFFFu - (unsigned)(keys[s] >> 32);
            if (__uint_as_float(sbits) > CLS_TH) word |= (1u << b);
        }
    }
    validw[w] = word;
}

// ---------------------------------------------------------------------------
// Kernel 4 (Phase A, whole-chip parallel): suppression bitmask.
// Block = 32 sorted rows x NWP word-columns. Row boxes staged into LDS with
// gfx1250 async global->LDS loads (ASYNCcnt). Rows processed in register
// chunks of 8 so LDS reads stay at 32 b128/thread. mask[i*NWP+w] bit b set
// iff j = 32w+b satisfies j>i, j<N, IoU(i,j) > 0.5.  Writes are coalesced.
// ---------------------------------------------------------------------------
__global__ __launch_bounds__(NWP) void mask_kernel(const float4* __restrict__ sboxes,
                                                   unsigned* __restrict__ mask,
                                                   int N) {
    __shared__ float4 rb[32];
    const int w = threadIdx.x;
    const int rowbase = blockIdx.x << 5;
    const int jbase = w << 5;

    if (threadIdx.x < 32) {                       // one wave32 issues the stage
        int i = rowbase + threadIdx.x;
        if (i < N) {
            u64 gaddr = (u64)(const void*)(sboxes + i);
            unsigned lds_addr = (unsigned)(size_t)(void*)(rb + threadIdx.x);
            asm volatile("global_load_async_to_lds_b128 %0, %1, off"
                         :: "v"(lds_addr), "v"(gaddr) : "memory");
        }
    }
    asm volatile("s_wait_asynccnt 0" ::: "memory");
    __syncthreads();

    for (int rc = 0; rc < 4; ++rc) {              // row chunks of 8
        const int r0 = rc << 3;
        unsigned wrd[8];
        float4 br[8];
        float ar[8];
#pragma unroll
        for (int r = 0; r < 8; ++r) {
            wrd[r] = 0u;
            br[r] = rb[r0 + r];
            ar[r] = (br[r].z - br[r].x) * (br[r].w - br[r].y);
        }
        // skip fully-below-diagonal chunks (max j <= min i -> all bits 0)
        if (jbase + 31 > rowbase + r0) {
            for (int b = 0; b < 32; ++b) {
                const int j = jbase + b;
                if (j >= N) break;                // per-thread tail guard
                float4 bj = sboxes[j];            // hot in WGP$/L2
                float areaj = (bj.z - bj.x) * (bj.w - bj.y);
#pragma unroll
                for (int r = 0; r < 8; ++r) {
                    const int i = rowbase + r0 + r;
                    float xx1 = fmaxf(br[r].x, bj.x);
                    float yy1 = fmaxf(br[r].y, bj.y);
                    float xx2 = fminf(br[r].z, bj.z);
                    float yy2 = fminf(br[r].w, bj.w);
                    float inter = fmaxf(xx2 - xx1, 0.0f) * fmaxf(yy2 - yy1, 0.0f);
                    float iou = inter / (ar[r] + areaj - inter + EPS_);
                    if ((j > i) && (iou > NMS_TH)) wrd[r] |= (1u << b);
                }
            }
        }
#pragma unroll
        for (int r = 0; r < 8; ++r) {
            const int i = rowbase + r0 + r;
            if (i < N) mask[(size_t)i * NWP + w] = wrd[r];  // coalesced over w
        }
    }
}

// ---------------------------------------------------------------------------
// Kernel 5 (Phase B): sequential NMS scan on ONE wave32 -- lockstep, so the
// LDS `remv` accumulator needs no barriers at all (DS ops are in-order per
// wave). 8-row software-pipelined prefetch (3 x uint4 per lane per row)
// hides L2 latency behind the OR work.
// ---------------------------------------------------------------------------
#define SCAN_D 8
__global__ __launch_bounds__(32) void scan_kernel(const unsigned* __restrict__ mask,
                                                  const unsigned* __restrict__ validw,
                                                  unsigned* __restrict__ keepw,
                                                  int N) {
    __shared__ unsigned remv[NWP];
    __shared__ unsigned vw[NWP];
    const int lane = threadIdx.x;

    for (int w = lane; w < NWP; w += 32) { remv[w] = 0u; vw[w] = validw[w]; }

    uint4 buf[SCAN_D][3];
#pragma unroll
    for (int d = 0; d < SCAN_D; ++d) {
        if (d < N) {
            const uint4* row = (const uint4*)(mask + (size_t)d * NWP);
            buf[d][0] = row[lane];
            buf[d][1] = row[lane + 32];
            buf[d][2] = row[lane + 64];
        }
    }

    for (int ib = 0; ib < N; ib += SCAN_D) {
#pragma unroll
        for (int d = 0; d < SCAN_D; ++d) {
            const int i = ib + d;
            uint4 r0 = buf[d][0], r1 = buf[d][1], r2 = buf[d][2];
            const int ip = i + SCAN_D;
            if (ip < N) {                         // uniform: prefetch next row
                const uint4* row = (const uint4*)(mask + (size_t)ip * NWP);
                buf[d][0] = row[lane];
                buf[d][1] = row[lane + 32];
                buf[d][2] = row[lane + 64];
            }
            const int wi = i >> 5;
            unsigned kept = (vw[wi] & ~remv[wi]) >> (i & 31);
            if (kept & 1u) {                      // uniform branch
                const int c0 = lane << 2;
                remv[c0 + 0] |= r0.x; remv[c0 + 1] |= r0.y;
                remv[c0 + 2] |= r0.z; remv[c0 + 3] |= r0.w;
                const int c1 = 128 + (lane << 2);
                remv[c1 + 0] |= r1.x; remv[c1 + 1] |= r1.y;
                remv[c1 + 2] |= r1.z; remv[c1 + 3] |= r1.w;
                const int c2 = 256 + (lane << 2);
                remv[c2 + 0] |= r2.x; remv[c2 + 1] |= r2.y;
                remv[c2 + 2] |= r2.z; remv[c2 + 3] |= r2.w;
            }
        }
    }

    for (int w = lane; w < NWP; w += 32) keepw[w] = vw[w] & ~remv[w];
}

// ---------------------------------------------------------------------------
// Kernel 6: scatter [x1,y1,x2,y2,score]*keep back to original index order.
// ---------------------------------------------------------------------------
__global__ void scatter_kernel(const u64* __restrict__ keys,
                               const float4* __restrict__ sboxes,
                               const unsigned* __restrict__ keepw,
                               float* __restrict__ out, int N) {
    int s = blockIdx.x * blockDim.x + threadIdx.x;
    if (s >= N) return;
    u64 key = keys[s];
    unsigned idx   = (unsigned)key;
    unsigned sbits = 0xFFFFFFFFu - (unsigned)(key >> 32);
    float score    = __uint_as_float(sbits);
    float m = ((keepw[s >> 5] >> (s & 31)) & 1u) ? 1.0f : 0.0f;
    float4 b = sboxes[s];
    float* o = out + (size_t)idx * 5;
    o[0] = b.x * m; o[1] = b.y * m; o[2] = b.z * m; o[3] = b.w * m; o[4] = score * m;
}

// ---------------------------------------------------------------------------
// Fallback (ws too small for the mask): fused LDS-resident barriered greedy
// NMS + scatter, boxes gathered via async load-to-LDS. 192KB+ LDS.
// ---------------------------------------------------------------------------
__global__ __launch_bounds__(1024) void nms_greedy_kernel(const u64* __restrict__ keys,
                                                          const float4* __restrict__ boxes,
                                                          float* __restrict__ out, int N) {
    extern __shared__ char dyn_smem[];
    float4*   sb    = (float4*)dyn_smem;
    unsigned* keepw = (unsigned*)(dyn_smem + (size_t)N * 16);
    const int nw = (N + 31) >> 5;

    for (int w = threadIdx.x; w < nw; w += blockDim.x) keepw[w] = 0u;
    __syncthreads();

    for (int s = threadIdx.x; s < N; s += blockDim.x) {
        u64 key = keys[s];
        unsigned idx   = (unsigned)key;
        unsigned sbits = 0xFFFFFFFFu - (unsigned)(key >> 32);
        u64 gaddr = (u64)(const void*)(boxes + idx);
        unsigned lds_addr = (unsigned)(size_t)(void*)(sb + s);
        asm volatile("global_load_async_to_lds_b128 %0, %1, off"
                     :: "v"(lds_addr), "v"(gaddr) : "memory");
        if (__uint_as_float(sbits) > CLS_TH) atomicOr(&keepw[s >> 5], 1u << (s & 31));
    }
    asm volatile("s_wait_asynccnt 0" ::: "memory");
    __syncthreads();

    for (int i = 0; i < N; ++i) {
        __syncthreads();
        if (!((keepw[i >> 5] >> (i & 31)) & 1u)) continue;
        float4 bi = sb[i];
        float areai = (bi.z - bi.x) * (bi.w - bi.y);
        for (int j = i + 1 + (int)threadIdx.x; j < N; j += blockDim.x) {
            if (!((keepw[j >> 5] >> (j & 31)) & 1u)) continue;
            float4 bj = sb[j];
            float xx1 = fmaxf(bi.x, bj.x), yy1 = fmaxf(bi.y, bj.y);
            float xx2 = fminf(bi.z, bj.z), yy2 = fminf(bi.w, bj.w);
            float inter = fmaxf(xx2 - xx1, 0.0f) * fmaxf(yy2 - yy1, 0.0f);
            float areaj = (bj.z - bj.x) * (bj.w - bj.y);
            if (inter / (areai + areaj - inter + EPS_) > NMS_TH)
                atomicAnd(&keepw[j >> 5], ~(1u << (j & 31)));
        }
    }
    __syncthreads();

    for (int s = threadIdx.x; s < N; s += blockDim.x) {
        u64 key = keys[s];
        unsigned idx   = (unsigned)key;
        unsigned sbits = 0xFFFFFFFFu - (unsigned)(key >> 32);
        float score    = __uint_as_float(sbits);
        float m = ((keepw[s >> 5] >> (s & 31)) & 1u) ? 1.0f : 0.0f;
        float4 b = sb[s];
        float* o = out + (size_t)idx * 5;
        o[0] = b.x * m; o[1] = b.y * m; o[2] = b.z * m; o[3] = b.w * m; o[4] = score * m;
    }
}

// ---------------------------------------------------------------------------
extern "C" void kernel_launch(void* const* d_in, const int* in_sizes, int n_in,
                              void* d_out, int out_size, void* d_ws, size_t ws_size,
                              hipStream_t stream) {
    (void)n_in; (void)out_size;
    const float* loc = (const float*)d_in[0];
    const float* cls = (const float*)d_in[1];
    const int*   pW  = (const int*)d_in[2];
    const int*   pH  = (const int*)d_in[3];

    int N = in_sizes[0] / 4;            // loc_preds is (N,4)
    int M = 1; while (M < N) M <<= 1;   // 16384 for N=12288

    auto al = [](size_t x) { return (x + 255) & ~(size_t)255; };
    size_t offBoxes = 0;
    size_t offKeys  = al(offBoxes + (size_t)N * 16);
    size_t offSB    = al(offKeys + (size_t)M * 8);
    size_t offValid = al(offSB + (size_t)N * 16);
    size_t offKeep  = al(offValid + (size_t)NWP * 4);
    size_t offMask  = al(offKeep + (size_t)NWP * 4);
    size_t need     = offMask + (size_t)N * NWP * 4;

    char* ws = (char*)d_ws;
    float4* boxes  = (float4*)(ws + offBoxes);
    u64*    keys   = (u64*)(ws + offKeys);

    decode_kernel<<<(N + 255) / 256, 256, 0, stream>>>(loc, cls, pW, pH, boxes, keys, N);
    sort_kernel<<<1, 1024, (size_t)M * sizeof(u64), stream>>>(keys, N, M);

    bool maskPath = (ws_size >= need) && (N % 32 == 0) &&
                    (((N + 31) / 32) <= NWP) && (N >= SCAN_D);
    if (maskPath) {
        float4*   sboxes = (float4*)(ws + offSB);
        unsigned* validw = (unsigned*)(ws + offValid);
        unsigned* keepw  = (unsigned*)(ws + offKeep);
        unsigned* mask   = (unsigned*)(ws + offMask);

        gather_kernel<<<(N + 255) / 256, 256, 0, stream>>>(keys, boxes, sboxes, N);
        validw_kernel<<<(NWP + 127) / 128, 128, 0, stream>>>(keys, validw, N);
        mask_kernel<<<N / 32, NWP, 0, stream>>>(sboxes, mask, N);
        scan_kernel<<<1, 32, 0, stream>>>(mask, validw, keepw, N);
        scatter_kernel<<<(N + 255) / 256, 256, 0, stream>>>(keys, sboxes, keepw,
                                                            (float*)d_out, N);
    } else {
        int nw = (N + 31) / 32;
        size_t lds = (size_t)N * 16 + (size_t)nw * 4;
        nms_greedy_kernel<<<1, 1024, lds, stream>>>(keys, boxes, (float*)d_out, N);
    }
}